// MultiLayerDGCN_2843268350771
// MI455X (gfx1250) — compile-verified
//
#include <hip/hip_runtime.h>

// CDNA5 fp32 WMMA fragment types
typedef __attribute__((ext_vector_type(2))) float v2f;
typedef __attribute__((ext_vector_type(8))) float v8f;
typedef __attribute__((ext_vector_type(4))) float v4f;

static __device__ __forceinline__ void atomic_add_f32(float* p, float v) {
    // lowers to global_atomic_add_f32 on gfx1250 (L2 RMW, no CAS loop)
    unsafeAtomicAdd(p, v);
}

// ---------------- graph preprocessing ----------------

__global__ void k_zero(float* p, int n) {
    int i = blockIdx.x * blockDim.x + threadIdx.x;
    if (i < n) p[i] = 0.0f;
}

__global__ void k_count_deg(const long long* __restrict__ dst, float* __restrict__ deg, int E) {
    int e = blockIdx.x * blockDim.x + threadIdx.x;
    if (e < E) atomic_add_f32(&deg[(int)dst[e]], 1.0f);
}

// deg -> deg^{-1/2} in place (self loop contributes +1, so deg+1 > 0 always)
__global__ void k_dis(float* __restrict__ deg, int n) {
    int i = blockIdx.x * blockDim.x + threadIdx.x;
    if (i < n) deg[i] = rsqrtf(deg[i] + 1.0f);
}

__global__ void k_norm(const long long* __restrict__ src, const long long* __restrict__ dst,
                       const float* __restrict__ dis, float* __restrict__ norm, int E) {
    int e = blockIdx.x * blockDim.x + threadIdx.x;
    if (e < E) norm[e] = dis[(int)src[e]] * dis[(int)dst[e]];
}

// ---------------- x_proj = x @ Wp + bp  (out width 4: plain VALU) ----------------

__global__ void k_proj(const float* __restrict__ x, const float* __restrict__ Wp,
                       const float* __restrict__ bp, float* __restrict__ xp, int N) {
    int t = blockIdx.x * blockDim.x + threadIdx.x;
    if (t >= N * 4) return;
    int i = t >> 2, j = t & 3;
    const float* xr = x + (size_t)i * 128;
    float s = bp[j];
#pragma unroll 8
    for (int k = 0; k < 128; ++k) s = fmaf(xr[k], Wp[k * 4 + j], s);
    xp[t] = s;
}

// ---------------- dense GEMM via V_WMMA_F32_16X16X4_F32 ----------------
// C[M,OUTC] = A[M,K] @ B[K,OUTC], row-major. grid.x = M/16 strips, 256 thr = 8 waves,
// wave w owns column tiles w, w+8, ... K multiple of 4, OUTC multiple of 16.
__global__ void k_gemm_wmma(const float* __restrict__ A, const float* __restrict__ B,
                            float* __restrict__ C, int K, int OUTC) {
    const int wave = threadIdx.x >> 5;
    const int lane = threadIdx.x & 31;
    const int half = lane >> 4;     // 0: K pair {k,k+1}; 1: {k+2,k+3}
    const int l    = lane & 15;     // A: row within tile; B/D: column within tile
    const int row  = blockIdx.x * 16 + l;
    const float* arow = A + (size_t)row * K + 2 * half;
    const int tiles = OUTC >> 4;
    for (int ct = wave; ct < tiles; ct += 8) {
        v8f c = {};
        const int col = ct * 16 + l;
        for (int k = 0; k < K; k += 4) {
            v2f a, b;
            const float* ap = arow + k;                               // 8B aligned
            a[0] = ap[0]; a[1] = ap[1];
            const float* bp = B + (size_t)(k + 2 * half) * OUTC + col;
            b[0] = bp[0]; b[1] = bp[OUTC];
            c = __builtin_amdgcn_wmma_f32_16x16x4_f32(
                    false, a, false, b, (short)0, c, false, false);
        }
        // D layout: VGPR r -> M = r + 8*half, N = l
        float* crow = C + (size_t)(blockIdx.x * 16 + 8 * half) * OUTC + ct * 16 + l;
#pragma unroll
        for (int r = 0; r < 8; ++r) crow[(size_t)r * OUTC] = c[r];
    }
}

// ---------------- aggregation (self loop folded into init) ----------------

// Ccat[i, 0:128] = H[i]*dis[i]^2 ; Ccat[i, 128:132] = xp[i]  (row stride 132)
__global__ void k_selfinit_cat(float* __restrict__ Cc, const float* __restrict__ H,
                               const float* __restrict__ xp, const float* __restrict__ dis, int N) {
    int t = blockIdx.x * blockDim.x + threadIdx.x;
    if (t >= N * 132) return;
    int i = t / 132, f = t - i * 132;
    float d = dis[i];
    Cc[t] = (f < 128) ? H[(size_t)i * 128 + f] * d * d : xp[i * 4 + (f - 128)];
}

// overwrite only the first 128 columns (xp columns persist from previous layer)
__global__ void k_selfinit128(float* __restrict__ Cc, const float* __restrict__ H,
                              const float* __restrict__ dis, int N) {
    int t = blockIdx.x * blockDim.x + threadIdx.x;
    if (t >= N * 128) return;
    int i = t >> 7, f = t & 127;
    float d = dis[i];
    Cc[(size_t)i * 132 + f] = H[t] * d * d;
}

// out[i*64+f] = H[i*64+f]*dis[i]^2 + b2[f]  (final layer, bias folded, no relu)
__global__ void k_init_out(const float* __restrict__ H, const float* __restrict__ dis,
                           const float* __restrict__ b2, float* __restrict__ out, int N) {
    int t = blockIdx.x * blockDim.x + threadIdx.x;
    if (t >= N * 64) return;
    int i = t >> 6, f = t & 63;
    float d = dis[i];
    out[t] = H[t] * d * d + b2[f];
}

// scatter-add messages: out[dst, f] += H[src, f] * norm[e].
// lfq = log2(F/4); one thread handles 4 features of one edge (float4 gather, 4 f32 atomics).
__global__ void k_scatter(const long long* __restrict__ src, const long long* __restrict__ dst,
                          const float* __restrict__ norm, const float* __restrict__ H, int HS,
                          float* __restrict__ out, int OS, int E, int lfq) {
    int t = blockIdx.x * blockDim.x + threadIdx.x;
    int e = t >> lfq;
    if (e >= E) return;
    int f4 = (t - (e << lfq)) << 2;
    float n = norm[e];
    const float* hp = H + (size_t)((int)src[e]) * HS + f4;   // 16B aligned
    v4f m = *(const v4f*)hp;
    float* op = out + (size_t)((int)dst[e]) * OS + f4;
    atomic_add_f32(op + 0, m[0] * n);
    atomic_add_f32(op + 1, m[1] * n);
    atomic_add_f32(op + 2, m[2] * n);
    atomic_add_f32(op + 3, m[3] * n);
}

// ---------------- bias + relu on the 128-wide part of Ccat ----------------

__global__ void k_bias_relu(float* __restrict__ Cc, const float* __restrict__ b, int N) {
    int t = blockIdx.x * blockDim.x + threadIdx.x;
    if (t >= N * 128) return;
    int i = t >> 7, f = t & 127;
    size_t idx = (size_t)i * 132 + f;
    float v = Cc[idx] + b[f];
    Cc[idx] = v > 0.0f ? v : 0.0f;
}

static inline int cdiv(int a, int b) { return (a + b - 1) / b; }

extern "C" void kernel_launch(void* const* d_in, const int* in_sizes, int n_in,
                              void* d_out, int out_size, void* d_ws, size_t ws_size,
                              hipStream_t stream) {
    const long long* ei = (const long long*)d_in[0];
    const int E = in_sizes[0] / 2;
    const float* x  = (const float*)d_in[1];
    const int N = in_sizes[1] / 128;            // 100000, multiple of 16
    const float* Wp = (const float*)d_in[2];
    const float* bp = (const float*)d_in[3];
    const float* W0 = (const float*)d_in[4];
    const float* b0 = (const float*)d_in[5];
    const float* W1 = (const float*)d_in[6];
    const float* b1 = (const float*)d_in[7];
    const float* W2 = (const float*)d_in[8];
    const float* b2 = (const float*)d_in[9];
    float* out = (float*)d_out;

    const long long* e_src = ei;
    const long long* e_dst = ei + E;

    float* ws   = (float*)d_ws;
    float* dis  = ws;                               // N   (deg, then deg^-1/2)
    float* nrm  = dis + N;                          // E
    float* xp   = nrm + E;                          // 4N
    float* H    = xp + (size_t)4 * N;               // 128N (GEMM outputs)
    float* Cc   = H + (size_t)128 * N;              // 132N (concat / aggregation buffer)

    const int B = 256;

    // graph norm
    k_zero      <<<cdiv(N, B), B, 0, stream>>>(dis, N);
    k_count_deg <<<cdiv(E, B), B, 0, stream>>>(e_dst, dis, E);
    k_dis       <<<cdiv(N, B), B, 0, stream>>>(dis, N);
    k_norm      <<<cdiv(E, B), B, 0, stream>>>(e_src, e_dst, dis, nrm, E);

    // x_proj
    k_proj <<<cdiv(N * 4, B), B, 0, stream>>>(x, Wp, bp, xp, N);

    // ---- layer 0: H = x @ W0 ; aggregate into Cc[:, :128]; +b0, relu; Cc[:,128:132]=xp
    k_gemm_wmma   <<<N / 16, B, 0, stream>>>(x, W0, H, 128, 128);
    k_selfinit_cat<<<cdiv(N * 132, B), B, 0, stream>>>(Cc, H, xp, dis, N);
    k_scatter     <<<cdiv(E * 32, B), B, 0, stream>>>(e_src, e_dst, nrm, H, 128, Cc, 132, E, 5);
    k_bias_relu   <<<cdiv(N * 128, B), B, 0, stream>>>(Cc, b0, N);

    // ---- layer 1: H = Cc(132) @ W1 ; aggregate; +b1, relu (xp columns persist)
    k_gemm_wmma   <<<N / 16, B, 0, stream>>>(Cc, W1, H, 132, 128);
    k_selfinit128 <<<cdiv(N * 128, B), B, 0, stream>>>(Cc, H, dis, N);
    k_scatter     <<<cdiv(E * 32, B), B, 0, stream>>>(e_src, e_dst, nrm, H, 128, Cc, 132, E, 5);
    k_bias_relu   <<<cdiv(N * 128, B), B, 0, stream>>>(Cc, b1, N);

    // ---- layer 2: H = Cc(132) @ W2 (out 64); aggregate straight into d_out, b2 folded
    k_gemm_wmma <<<N / 16, B, 0, stream>>>(Cc, W2, H, 132, 64);
    k_init_out  <<<cdiv(N * 64, B), B, 0, stream>>>(H, dis, b2, out, N);
    k_scatter   <<<cdiv(E * 16, B), B, 0, stream>>>(e_src, e_dst, nrm, H, 64, out, 64, E, 4);
}